// TPUMBlock_24730421691119
// MI455X (gfx1250) — compile-verified
//
#include <hip/hip_runtime.h>
#include <hip/hip_bf16.h>
#include <math.h>

#define D_    768
#define H_    12
#define DH_   64
#define WIN_  256
#define DS_   256
#define MS_   16
#define B_    2
#define T_    2048
#define ROWS_ (B_ * T_)   /* 4096 */

typedef float v2f __attribute__((ext_vector_type(2)));
typedef float v8f __attribute__((ext_vector_type(8)));

__device__ __forceinline__ v8f wmma_f32(v2f a, v2f b, v8f c) {
  // V_WMMA_F32_16X16X4_F32 : D(16x16 f32) = A(16x4 f32) * B(4x16 f32) + C
  return __builtin_amdgcn_wmma_f32_16x16x4_f32(
      /*neg_a=*/false, a, /*neg_b=*/false, b,
      /*c_mod=*/(short)0, c, /*reuse_a=*/false, /*reuse_b=*/false);
}

// LDS byte offset of a __shared__ object: flat shared-aperture addresses map
// to LDS by truncation to the low 32 bits (ISA 10.2 aperture rules).
__device__ __forceinline__ uint32_t lds_off(const void* p) {
  return (uint32_t)(uintptr_t)p;
}

// Async 16-byte global -> LDS copy, tracked by ASYNCcnt (CDNA5 TDM-lite path).
__device__ __forceinline__ void async_b128(void* lds, const float* g) {
  asm volatile("global_load_async_to_lds_b128 %0, %1, off"
               :: "v"(lds_off(lds)), "v"(g) : "memory");
}
__device__ __forceinline__ void async_wait_prev() {   // keep newest 4 in flight
  asm volatile("s_wait_asynccnt 4" ::: "memory");
}
__device__ __forceinline__ void async_wait_all() {
  asm volatile("s_wait_asynccnt 0" ::: "memory");
}

// ---------------------------------------------------------------------------
// LayerNorm (+ optional 3-way router softmax).  One 256-thread block per row.
// ---------------------------------------------------------------------------
__global__ __launch_bounds__(256) void ln_router_kernel(
    const float* __restrict__ xin, const float* __restrict__ gw,
    const float* __restrict__ bw, const float* __restrict__ Wr,
    const float* __restrict__ br, float* __restrict__ xn,
    float* __restrict__ probs)
{
  __shared__ float r0[256], r1[256], r2[256];
  const int row = blockIdx.x;
  const int tid = threadIdx.x;
  const float* xr = xin + (size_t)row * D_;
  float lv[3];
  float s = 0.f, ss = 0.f;
#pragma unroll
  for (int i = 0; i < 3; ++i) {
    float v = xr[tid + i * 256];
    lv[i] = v; s += v; ss += v * v;
  }
  r0[tid] = s; r1[tid] = ss;
  __syncthreads();
  for (int off = 128; off > 0; off >>= 1) {
    if (tid < off) { r0[tid] += r0[tid + off]; r1[tid] += r1[tid + off]; }
    __syncthreads();
  }
  const float mean = r0[0] * (1.0f / D_);
  const float var  = r1[0] * (1.0f / D_) - mean * mean;
  const float rstd = rsqrtf(var + 1e-5f);
  __syncthreads();

  float p0 = 0.f, p1 = 0.f, p2 = 0.f;
#pragma unroll
  for (int i = 0; i < 3; ++i) {
    int c = tid + i * 256;
    float v = (lv[i] - mean) * rstd * gw[c] + bw[c];
    xn[(size_t)row * D_ + c] = v;
    if (Wr) { p0 += v * Wr[c * 3 + 0]; p1 += v * Wr[c * 3 + 1]; p2 += v * Wr[c * 3 + 2]; }
  }
  if (Wr) {
    r0[tid] = p0; r1[tid] = p1; r2[tid] = p2;
    __syncthreads();
    for (int off = 128; off > 0; off >>= 1) {
      if (tid < off) { r0[tid] += r0[tid + off]; r1[tid] += r1[tid + off]; r2[tid] += r2[tid + off]; }
      __syncthreads();
    }
    if (tid == 0) {
      float a = r0[0] + br[0], b = r1[0] + br[1], c = r2[0] + br[2];
      float mx = fmaxf(a, fmaxf(b, c));
      float ea = expf(a - mx), eb = expf(b - mx), ec = expf(c - mx);
      float inv = 1.0f / (ea + eb + ec);
      probs[row * 3 + 0] = ea * inv;
      probs[row * 3 + 1] = eb * inv;
      probs[row * 3 + 2] = ec * inv;
    }
  }
}

// ---------------------------------------------------------------------------
// fp32 WMMA GEMM:  C[M,N] = act(A[M,K] @ W[K,N] + bias) (+ Res)
// 128x128 block tile, 8 wave32s, each wave owns a 32x64 strip (acc[2][4]).
// K staged 16-wide through double-buffered LDS filled by async B128 copies.
// Requires N % 128 == 0 and K % 16 == 0 (true for all uses here); M guarded
// (out-of-range A rows are clamped to row 0 and their C rows never stored).
// ---------------------------------------------------------------------------
__global__ __launch_bounds__(256) void gemm_kernel(
    const float* __restrict__ A, const float* __restrict__ W,
    const float* __restrict__ bias, const float* __restrict__ Res,
    float* __restrict__ C, int M, int N, int K, int act)
{
  __shared__ __align__(16) float As[2][128][28];   // 16 cols used, row=112B (16B-aligned, conflict-free)
  __shared__ __align__(16) float Bs[2][16][132];   // 128 cols used, row=528B

  const int tid   = threadIdx.x;
  const int lane  = tid & 31;
  const int wave  = tid >> 5;          // 0..7
  const int mrow0 = (wave & 3) * 32;   // two 16-row subtiles: mrow0, mrow0+16
  const int ncol0 = (wave >> 2) * 64;  // four 16-col subtiles
  const int n0    = blockIdx.x * 128;
  const int m0    = blockIdx.y * 128;
  const int lr    = lane & 15;
  const int koff  = (lane >> 4) * 2;

  // per-thread async-copy coordinates
  const int a_row = tid >> 1;          // 0..127
  const int a_kp  = (tid & 1) * 8;     // 0 or 8
  int a_gr = m0 + a_row; if (a_gr >= M) a_gr = 0;   // clamp: row unused in epilogue
  const int b_kr = tid >> 4;           // 0..15
  const int b_c  = (tid & 15) * 8;     // 0..120 step 8

  const v8f vzero = {0.f,0.f,0.f,0.f,0.f,0.f,0.f,0.f};
  v8f acc[2][4];
#pragma unroll
  for (int s = 0; s < 2; ++s)
#pragma unroll
    for (int n = 0; n < 4; ++n) acc[s][n] = vzero;

  // stage k-chunk `k0` into buffer `buf` (4 async B128 issues per wave)
  auto stage = [&](int buf, int k0) {
    const float* ga = A + (size_t)a_gr * K + k0 + a_kp;
    async_b128(&As[buf][a_row][a_kp],     ga);
    async_b128(&As[buf][a_row][a_kp + 4], ga + 4);
    const float* gb = W + (size_t)(k0 + b_kr) * N + n0 + b_c;
    async_b128(&Bs[buf][b_kr][b_c],     gb);
    async_b128(&Bs[buf][b_kr][b_c + 4], gb + 4);
  };

  int buf = 0;
  stage(0, 0);
  for (int k0 = 0; k0 < K; k0 += 16) {
    const bool more = (k0 + 16) < K;
    if (more) stage(buf ^ 1, k0 + 16);   // prefetch next tile into other buffer
    if (more) async_wait_prev(); else async_wait_all();
    __syncthreads();

#pragma unroll
    for (int kk = 0; kk < 4; ++kk) {
      v2f a0, a1;
      a0.x = As[buf][mrow0 + lr][kk * 4 + koff];
      a0.y = As[buf][mrow0 + lr][kk * 4 + koff + 1];
      a1.x = As[buf][mrow0 + 16 + lr][kk * 4 + koff];
      a1.y = As[buf][mrow0 + 16 + lr][kk * 4 + koff + 1];
#pragma unroll
      for (int ns = 0; ns < 4; ++ns) {
        int col = ncol0 + ns * 16 + lr;
        v2f b;
        b.x = Bs[buf][kk * 4 + koff][col];
        b.y = Bs[buf][kk * 4 + koff + 1][col];
        acc[0][ns] = wmma_f32(a0, b, acc[0][ns]);
        acc[1][ns] = wmma_f32(a1, b, acc[1][ns]);
      }
    }
    __syncthreads();
    buf ^= 1;
  }

  const int rowoff = 8 * (lane >> 4);
#pragma unroll
  for (int s = 0; s < 2; ++s) {
#pragma unroll
    for (int ns = 0; ns < 4; ++ns) {
      int gc = n0 + ncol0 + ns * 16 + lr;
      float bb = bias ? bias[gc] : 0.0f;
#pragma unroll
      for (int r = 0; r < 8; ++r) {
        int gr = m0 + mrow0 + s * 16 + rowoff + r;
        if (gr < M) {
          float v = acc[s][ns][r] + bb;
          if (act == 1) v = 0.5f * v * (1.0f + erff(v * 0.70710678118654752f));  // exact GELU
          if (Res) v += Res[(size_t)gr * N + gc];
          C[(size_t)gr * N + gc] = v;
        }
      }
    }
  }
}

// ---------------------------------------------------------------------------
// Banded-causal flash attention (Path 1).  One wave32 per (b, h, 16-q tile).
// qkv layout: [B,T,3D], q at [0,D), k at [D,2D), v at [2D,3D); head slice 64.
// out: [B,T,D] with (h*64+d) packing (i.e. already the transposed-merged form).
// ---------------------------------------------------------------------------
__global__ __launch_bounds__(32) void attn_local_kernel(
    const float* __restrict__ qkv, float* __restrict__ out)
{
  __shared__ float Qs[16][68];
  __shared__ float Ks[16][68];
  __shared__ float Vs[16][68];
  __shared__ float Ps[16][16];
  __shared__ float alpha_s[16];
  __shared__ float linv_s[16];

  const int b = blockIdx.z, h = blockIdx.y, qt = blockIdx.x;
  const int qstart = qt * 16;
  const int lane = threadIdx.x;
  const int lr = lane & 15;
  const int koff = (lane >> 4) * 2;
  const size_t base = (size_t)b * T_;
  const v8f vzero = {0.f,0.f,0.f,0.f,0.f,0.f,0.f,0.f};

#pragma unroll
  for (int i = 0; i < 32; ++i) {          // Q tile 16x64, pre-scaled by dh^-0.5
    int idx = lane + i * 32;
    int r = idx >> 6, d = idx & 63;
    Qs[r][d] = qkv[(base + qstart + r) * (3 * D_) + h * 64 + d] * 0.125f;
  }

  float m_i = -__builtin_inff(), l_i = 0.0f;  // live in lanes 0..15
  v8f o[4];
#pragma unroll
  for (int n = 0; n < 4; ++n) o[n] = vzero;

  const int jt0 = (qstart >= WIN_) ? ((qstart - WIN_) >> 4) : 0;
  for (int jt = jt0; jt <= qt; ++jt) {
    const int jstart = jt * 16;
#pragma unroll
    for (int i = 0; i < 32; ++i) {        // K,V tiles 16x64
      int idx = lane + i * 32;
      int r = idx >> 6, d = idx & 63;
      size_t rb = (base + jstart + r) * (3 * D_) + h * 64 + d;
      Ks[r][d] = qkv[rb + D_];
      Vs[r][d] = qkv[rb + 2 * D_];
    }
    __syncthreads();

    v8f s = vzero;                        // S = Q * K^T  (16 x f32 WMMAs)
#pragma unroll
    for (int kk = 0; kk < 16; ++kk) {
      v2f a;  a.x = Qs[lr][kk * 4 + koff];  a.y = Qs[lr][kk * 4 + koff + 1];
      v2f bf; bf.x = Ks[lr][kk * 4 + koff]; bf.y = Ks[lr][kk * 4 + koff + 1];
      s = wmma_f32(a, bf, s);
    }
#pragma unroll
    for (int r = 0; r < 8; ++r) Ps[r + 8 * (lane >> 4)][lr] = s[r];
    __syncthreads();

    if (lane < 16) {                      // row-wise online softmax
      const int gi = qstart + lane;
      float mx = m_i;
      float sv[16];
#pragma unroll
      for (int j = 0; j < 16; ++j) {
        int gj = jstart + j;
        bool ok = (gj <= gi) && (gj >= gi - WIN_);
        float v = ok ? Ps[lane][j] : -__builtin_inff();
        sv[j] = v; mx = fmaxf(mx, v);
      }
      float al = expf(m_i - mx);
      float sum = 0.f;
#pragma unroll
      for (int j = 0; j < 16; ++j) {
        float p = expf(sv[j] - mx);
        Ps[lane][j] = p; sum += p;
      }
      l_i = l_i * al + sum; m_i = mx; alpha_s[lane] = al;
    }
    __syncthreads();

    float al8[8];
#pragma unroll
    for (int r = 0; r < 8; ++r) al8[r] = alpha_s[r + 8 * (lane >> 4)];
#pragma unroll
    for (int n = 0; n < 4; ++n)
#pragma unroll
      for (int r = 0; r < 8; ++r) o[n][r] *= al8[r];

#pragma unroll
    for (int n = 0; n < 4; ++n)           // O += P * V
#pragma unroll
      for (int kk = 0; kk < 4; ++kk) {
        v2f a;  a.x = Ps[lr][kk * 4 + koff];          a.y = Ps[lr][kk * 4 + koff + 1];
        v2f bf; bf.x = Vs[kk * 4 + koff][n * 16 + lr]; bf.y = Vs[kk * 4 + koff + 1][n * 16 + lr];
        o[n] = wmma_f32(a, bf, o[n]);
      }
    __syncthreads();
  }

  if (lane < 16) linv_s[lane] = 1.0f / l_i;
  __syncthreads();
  float li8[8];
#pragma unroll
  for (int r = 0; r < 8; ++r) li8[r] = linv_s[r + 8 * (lane >> 4)];
#pragma unroll
  for (int n = 0; n < 4; ++n)
#pragma unroll
    for (int r = 0; r < 8; ++r) {
      int row = 8 * (lane >> 4) + r;
      out[(base + qstart + row) * D_ + h * 64 + n * 16 + lr] = o[n][r] * li8[r];
    }
}

// ---------------------------------------------------------------------------
// Path 2: diagonal linear recurrence.  512 threads, one per (b, state dim).
// h overwrites u in place; s_next written to output.
// ---------------------------------------------------------------------------
__global__ __launch_bounds__(256) void scan_kernel(
    float* __restrict__ u, const float* __restrict__ s0,
    const float* __restrict__ lam, const float* __restrict__ gam,
    float* __restrict__ s_out)
{
  const int idx = blockIdx.x * 256 + threadIdx.x;  // 0..511
  const int b = idx / DS_;
  const int d = idx % DS_;
  const float dec = 1.0f / (1.0f + expf(-lam[d]));
  const float gn  = gam[d];
  float s = s0[idx];
  for (int t = 0; t < T_; ++t) {
    size_t p = ((size_t)b * T_ + t) * DS_ + d;
    s = dec * s + gn * u[p];
    u[p] = s;
  }
  s_out[idx] = s;
}

// ---------------------------------------------------------------------------
// Path 3a: global-memory MHA read over MS=16 slots.  One block per (b,t).
// ---------------------------------------------------------------------------
__global__ __launch_bounds__(256) void gmem_read_kernel(
    const float* __restrict__ qg, const float* __restrict__ kg,
    const float* __restrict__ vg, float* __restrict__ ygp)
{
  __shared__ float sc[H_][MS_];
  const int bt = blockIdx.x;
  const int b = bt / T_;
  const int tid = threadIdx.x;
  const float* qrow = qg + (size_t)bt * D_;

  if (tid < H_ * MS_) {
    int h = tid >> 4, m = tid & 15;
    const float* krow = kg + ((size_t)b * MS_ + m) * D_ + h * 64;
    const float* q2 = qrow + h * 64;
    float acc = 0.f;
    for (int d = 0; d < 64; ++d) acc += q2[d] * krow[d];
    sc[h][m] = acc * 0.125f;
  }
  __syncthreads();
  if (tid < H_) {
    float mx = -__builtin_inff();
    for (int m = 0; m < MS_; ++m) mx = fmaxf(mx, sc[tid][m]);
    float s = 0.f;
    for (int m = 0; m < MS_; ++m) { float e = expf(sc[tid][m] - mx); sc[tid][m] = e; s += e; }
    float inv = 1.0f / s;
    for (int m = 0; m < MS_; ++m) sc[tid][m] *= inv;
  }
  __syncthreads();
  for (int o = tid; o < D_; o += 256) {
    int h = o >> 6;
    float acc = 0.f;
    for (int m = 0; m < MS_; ++m) acc += sc[h][m] * vg[((size_t)b * MS_ + m) * D_ + o];
    ygp[(size_t)bt * D_ + o] = acc;
  }
}

// ---------------------------------------------------------------------------
// Path 3b: pooled memory write + sigmoid gate.  One block per (b, slot).
// ---------------------------------------------------------------------------
__global__ __launch_bounds__(256) void mem_write_kernel(
    const float* __restrict__ m_prev, const float* __restrict__ xn,
    const float* __restrict__ Wwg, const float* __restrict__ bwg,
    float* __restrict__ m_next)
{
  __shared__ float pool[T_];
  __shared__ float red[256];
  __shared__ float cands[D_];
  const int bm = blockIdx.x;
  const int b = bm / MS_;
  const int tid = threadIdx.x;
  const float* mp = m_prev + (size_t)bm * D_;
  const float invs = rsqrtf((float)D_);

  for (int t = tid; t < T_; t += 256) {
    const float* xr = xn + ((size_t)b * T_ + t) * D_;
    float acc = 0.f;
    for (int d = 0; d < D_; ++d) acc += mp[d] * xr[d];
    pool[t] = acc * invs;
  }
  __syncthreads();
  float mx = -__builtin_inff();
  for (int t = tid; t < T_; t += 256) mx = fmaxf(mx, pool[t]);
  red[tid] = mx; __syncthreads();
  for (int off = 128; off > 0; off >>= 1) {
    if (tid < off) red[tid] = fmaxf(red[tid], red[tid + off]);
    __syncthreads();
  }
  const float gmx = red[0];
  __syncthreads();
  float sm = 0.f;
  for (int t = tid; t < T_; t += 256) { float e = expf(pool[t] - gmx); pool[t] = e; sm += e; }
  __syncthreads();
  red[tid] = sm; __syncthreads();
  for (int off = 128; off > 0; off >>= 1) {
    if (tid < off) red[tid] += red[tid + off];
    __syncthreads();
  }
  const float inv = 1.0f / red[0];
  __syncthreads();
  for (int d = tid; d < D_; d += 256) {
    float acc = 0.f;
    for (int t = 0; t < T_; ++t) acc += pool[t] * xn[((size_t)b * T_ + t) * D_ + d];
    cands[d] = acc * inv;
  }
  __syncthreads();
  float gp = 0.f;
  for (int d = tid; d < D_; d += 256) gp += cands[d] * Wwg[d];
  red[tid] = gp; __syncthreads();
  for (int off = 128; off > 0; off >>= 1) {
    if (tid < off) red[tid] += red[tid + off];
    __syncthreads();
  }
  const float gate = 1.0f / (1.0f + expf(-(red[0] + bwg[0])));
  for (int d = tid; d < D_; d += 256)
    m_next[(size_t)bm * D_ + d] = (1.0f - gate) * mp[d] + gate * cands[d];
}

// ---------------------------------------------------------------------------
// Router mix + residual:  x1 = x + p0*y_l + p1*y_s + p2*y_g
// ---------------------------------------------------------------------------
__global__ __launch_bounds__(256) void mix_kernel(
    const float* __restrict__ x, const float* __restrict__ yl,
    const float* __restrict__ ys, const float* __restrict__ yg,
    const float* __restrict__ probs, float* __restrict__ x1)
{
  const int i = blockIdx.x * 256 + threadIdx.x;
  const int row = i / D_;
  const float p0 = probs[row * 3 + 0];
  const float p1 = probs[row * 3 + 1];
  const float p2 = probs[row * 3 + 2];
  x1[i] = x[i] + p0 * yl[i] + p1 * ys[i] + p2 * yg[i];
}

// ---------------------------------------------------------------------------
extern "C" void kernel_launch(void* const* d_in, const int* in_sizes, int n_in,
                              void* d_out, int out_size, void* d_ws, size_t ws_size,
                              hipStream_t stream) {
  (void)in_sizes; (void)n_in; (void)out_size; (void)ws_size;
  const float* x      = (const float*)d_in[0];
  const float* s_prev = (const float*)d_in[1];
  const float* m_prev = (const float*)d_in[2];
  const float* ln1_g  = (const float*)d_in[3];
  const float* ln1_b  = (const float*)d_in[4];
  const float* Wr     = (const float*)d_in[5];
  const float* br     = (const float*)d_in[6];
  const float* Wqkv   = (const float*)d_in[7];
  const float* bqkv   = (const float*)d_in[8];
  const float* Wo_l   = (const float*)d_in[9];
  const float* bo_l   = (const float*)d_in[10];
  const float* Win_s  = (const float*)d_in[11];
  const float* bin_s  = (const float*)d_in[12];
  const float* lam    = (const float*)d_in[13];
  const float* gamp   = (const float*)d_in[14];
  const float* Wout_s = (const float*)d_in[15];
  const float* bout_s = (const float*)d_in[16];
  const float* Wq_g   = (const float*)d_in[17];
  const float* bq_g   = (const float*)d_in[18];
  const float* Wk_g   = (const float*)d_in[19];
  const float* bk_g   = (const float*)d_in[20];
  const float* Wv_g   = (const float*)d_in[21];
  const float* bv_g   = (const float*)d_in[22];
  const float* Wo_g   = (const float*)d_in[23];
  const float* bo_g   = (const float*)d_in[24];
  const float* Wwg    = (const float*)d_in[25];
  const float* bwg    = (const float*)d_in[26];
  const float* ln2_g  = (const float*)d_in[27];
  const float* ln2_b  = (const float*)d_in[28];
  const float* W1     = (const float*)d_in[29];
  const float* b1     = (const float*)d_in[30];
  const float* W2     = (const float*)d_in[31];
  const float* b2     = (const float*)d_in[32];

  float* out   = (float*)d_out;                 // x2 [B,T,D]
  float* s_out = out + (size_t)ROWS_ * D_;      // s_next [B,DS]
  float* m_out = s_out + (size_t)B_ * DS_;      // m_next [B,MS,D]
  float* probs = m_out + (size_t)B_ * MS_ * D_; // probs [B,T,3]

  float* ws  = (float*)d_ws;
  float* XN  = ws;
  float* QKV = XN  + (size_t)ROWS_ * D_;
  float* ATT = QKV + (size_t)ROWS_ * 3 * D_;
  float* YL  = ATT + (size_t)ROWS_ * D_;
  float* U   = YL  + (size_t)ROWS_ * D_;
  float* YS  = U   + (size_t)ROWS_ * DS_;
  float* QG  = YS  + (size_t)ROWS_ * D_;
  float* KG  = QG  + (size_t)ROWS_ * D_;
  float* VG  = KG  + (size_t)B_ * MS_ * D_;
  float* YGP = VG  + (size_t)B_ * MS_ * D_;
  float* YG  = YGP + (size_t)ROWS_ * D_;
  float* X1  = YG  + (size_t)ROWS_ * D_;
  float* X1N = X1  + (size_t)ROWS_ * D_;
  float* HMID= X1N + (size_t)ROWS_ * D_;

  const dim3 blk(256);
  const int MB = ROWS_ / 128;   // 32 block-rows for the big GEMMs

  // LN1 + router probs
  ln_router_kernel<<<ROWS_, blk, 0, stream>>>(x, ln1_g, ln1_b, Wr, br, XN, probs);

  // Path 1: QKV -> banded flash attention -> Wo_l
  gemm_kernel<<<dim3(3 * D_ / 128, MB), blk, 0, stream>>>(
      XN, Wqkv, bqkv, nullptr, QKV, ROWS_, 3 * D_, D_, 0);
  attn_local_kernel<<<dim3(T_ / 16, H_, B_), dim3(32), 0, stream>>>(QKV, ATT);
  gemm_kernel<<<dim3(D_ / 128, MB), blk, 0, stream>>>(
      ATT, Wo_l, bo_l, nullptr, YL, ROWS_, D_, D_, 0);

  // Path 2: Win_s -> scan -> Wout_s
  gemm_kernel<<<dim3(DS_ / 128, MB), blk, 0, stream>>>(
      XN, Win_s, bin_s, nullptr, U, ROWS_, DS_, D_, 0);
  scan_kernel<<<(B_ * DS_) / 256, blk, 0, stream>>>(U, s_prev, lam, gamp, s_out);
  gemm_kernel<<<dim3(D_ / 128, MB), blk, 0, stream>>>(
      U, Wout_s, bout_s, nullptr, YS, ROWS_, D_, DS_, 0);

  // Path 3: memory read + gated write
  gemm_kernel<<<dim3(D_ / 128, MB), blk, 0, stream>>>(
      XN, Wq_g, bq_g, nullptr, QG, ROWS_, D_, D_, 0);
  gemm_kernel<<<dim3(D_ / 128, 1), blk, 0, stream>>>(
      m_prev, Wk_g, bk_g, nullptr, KG, B_ * MS_, D_, D_, 0);
  gemm_kernel<<<dim3(D_ / 128, 1), blk, 0, stream>>>(
      m_prev, Wv_g, bv_g, nullptr, VG, B_ * MS_, D_, D_, 0);
  gmem_read_kernel<<<ROWS_, blk, 0, stream>>>(QG, KG, VG, YGP);
  gemm_kernel<<<dim3(D_ / 128, MB), blk, 0, stream>>>(
      YGP, Wo_g, bo_g, nullptr, YG, ROWS_, D_, D_, 0);
  mem_write_kernel<<<B_ * MS_, blk, 0, stream>>>(m_prev, XN, Wwg, bwg, m_out);

  // Mix + LN2 + FFN (+ residual fused into final GEMM, writing x2)
  mix_kernel<<<(ROWS_ * D_) / 256, blk, 0, stream>>>(x, YL, YS, YG, probs, X1);
  ln_router_kernel<<<ROWS_, blk, 0, stream>>>(X1, ln2_g, ln2_b, nullptr, nullptr, X1N, nullptr);
  gemm_kernel<<<dim3(4 * D_ / 128, MB), blk, 0, stream>>>(
      X1N, W1, b1, nullptr, HMID, ROWS_, 4 * D_, D_, 1);
  gemm_kernel<<<dim3(D_ / 128, MB), blk, 0, stream>>>(
      HMID, W2, b2, X1, out, ROWS_, D_, 4 * D_, 0);
}